// Attention_60421599920240
// MI455X (gfx1250) — compile-verified
//
#include <hip/hip_runtime.h>

// ---------------------------------------------------------------------------
// Equivariant graph attention (MUL=8, 5 paths) for MI455X / gfx1250.
// Radial MLPs (the ~17.6 GFLOP hot spot) run on V_WMMA_F32_16X16X4_F32.
// ---------------------------------------------------------------------------

constexpr int kNodes = 25000;
constexpr int kEdges = 200000;

typedef __attribute__((ext_vector_type(2))) float v2f;
typedef __attribute__((ext_vector_type(8))) float v8f;

#define HS_LD 66   // padded row stride for H staging (bank-conflict free)
#define WB_LD 65   // padded row stride for w-tile staging
#define WAVES_B 4  // waves per block in the edge kernel

__device__ __forceinline__ v8f wmma4(v2f a, v2f b, v8f c) {
  // D = A(16x4,f32) * B(4x16,f32) + C(16x16,f32)
  return __builtin_amdgcn_wmma_f32_16x16x4_f32(false, a, false, b, (short)0, c,
                                               false, false);
}

// Fast reciprocal (v_rcp_f32, ~1 ulp) — avoids the ~15-op IEEE divide chain.
__device__ __forceinline__ float fast_rcp(float x) {
  return __builtin_amdgcn_rcpf(x);
}

// silu(t) = t * sigmoid(t), with approximate exp + rcp (both TRANS ops).
__device__ __forceinline__ float fast_silu(float t) {
  return t * fast_rcp(1.f + __expf(-t));
}

// ---------------------------------------------------------------------------
__global__ void eqa_zero_kernel(float* __restrict__ out, float* __restrict__ zc) {
  int i = blockIdx.x * blockDim.x + threadIdx.x;
  if (i < kNodes * 32) out[i] = 0.f;
  if (i < 2 * kNodes) zc[i] = 0.f;  // z_sum then cnt, contiguous
}

// ---------------------------------------------------------------------------
__global__ void eqa_q_kernel(const float* __restrict__ na,
                             const float* __restrict__ Wq0,
                             const float* __restrict__ Wq1,
                             float* __restrict__ qs, float* __restrict__ qv) {
  int n = blockIdx.x * blockDim.x + threadIdx.x;
  if (n >= kNodes) return;
  const float inv = 0.3535533905932738f;  // 1/sqrt(8)
  float nsv[8], nvv[8][3];
#pragma unroll
  for (int m = 0; m < 8; ++m) nsv[m] = na[n * 32 + m];
#pragma unroll
  for (int m = 0; m < 8; ++m)
#pragma unroll
    for (int c = 0; c < 3; ++c) nvv[m][c] = na[n * 32 + 8 + m * 3 + c];
#pragma unroll
  for (int k = 0; k < 8; ++k) {
    float a = 0.f;
#pragma unroll
    for (int m = 0; m < 8; ++m) a += nsv[m] * Wq0[m * 8 + k];
    qs[n * 8 + k] = a * inv;
  }
#pragma unroll
  for (int k = 0; k < 8; ++k)
#pragma unroll
    for (int c = 0; c < 3; ++c) {
      float a = 0.f;
#pragma unroll
      for (int m = 0; m < 8; ++m) a += nvv[m][c] * Wq1[m * 8 + k];
      qv[n * 24 + k * 3 + c] = a * inv;
    }
}

// ---------------------------------------------------------------------------
// Fused per-edge kernel: radial MLPs (WMMA) + tensor product + logits.
// Each wave owns 32 consecutive edges (two 16-row WMMA tiles).
// ---------------------------------------------------------------------------
__global__ __launch_bounds__(WAVES_B * 32)
void eqa_edge_kernel(const float* __restrict__ node_attr,
                     const int* __restrict__ edge_index,
                     const float* __restrict__ edge_attr,
                     const float* __restrict__ edge_sh,
                     const float* __restrict__ kW1, const float* __restrict__ kW2,
                     const float* __restrict__ vW1, const float* __restrict__ vW2,
                     const float* __restrict__ Wd0, const float* __restrict__ Wd1,
                     const float* __restrict__ qs, const float* __restrict__ qv,
                     float* __restrict__ zsum, float* __restrict__ cnt,
                     float* __restrict__ expbuf, float* __restrict__ vedge) {
  extern __shared__ float smem[];
  const int lane = threadIdx.x & 31;
  const int wid = threadIdx.x >> 5;
  float* Hs = smem + wid * (32 * HS_LD + 32 * WB_LD);  // H: 32 edges x 64
  float* wb = Hs + 32 * HS_LD;                         // w-tile: 32 edges x 64
  const int gw = blockIdx.x * WAVES_B + wid;
  const int e0 = gw * 32;
  if (e0 >= kEdges) return;  // wave-uniform: EXEC stays all-ones for WMMA

  const int half = lane >> 4;   // lane half selects K-pair (ISA A/B layout)
  const int l16 = lane & 15;
  const int koff = half * 2;

  // ------- per-lane (lane == edge) gathers for the tensor product ---------
  const int eL = e0 + lane;
  const int src = edge_index[eL];
  const int dst = edge_index[kEdges + eL];
  const float s0 = edge_sh[eL * 4 + 0];
  const float s1x = edge_sh[eL * 4 + 1];
  const float s1y = edge_sh[eL * 4 + 2];
  const float s1z = edge_sh[eL * 4 + 3];
  float xs[8], xv[8][3];
#pragma unroll
  for (int m = 0; m < 8; ++m) xs[m] = node_attr[src * 32 + m];
#pragma unroll
  for (int m = 0; m < 8; ++m) {
    xv[m][0] = node_attr[src * 32 + 8 + m * 3 + 0];
    xv[m][1] = node_attr[src * 32 + 8 + m * 3 + 1];
    xv[m][2] = node_attr[src * 32 + 8 + m * 3 + 2];
  }

  float ts[2][8];      // scalar-channel accumulators (phase 0 = k, 1 = v)
  float tvv[2][8][3];  // vector-channel accumulators
#pragma unroll
  for (int ph = 0; ph < 2; ++ph)
#pragma unroll
    for (int n = 0; n < 8; ++n) {
      ts[ph][n] = 0.f;
      tvv[ph][n][0] = 0.f; tvv[ph][n][1] = 0.f; tvv[ph][n][2] = 0.f;
    }

  for (int phase = 0; phase < 2; ++phase) {
    const float* W1 = phase ? vW1 : kW1;
    const float* W2 = phase ? vW2 : kW2;

    // ---- GEMM1: H = silu(edge_attr @ W1 / 4), 16x16x16 per M-tile --------
#pragma unroll
    for (int mt = 0; mt < 2; ++mt) {
      const float* arow = edge_attr + (e0 + mt * 16 + l16) * 16;
      v2f af[4];
#pragma unroll
      for (int kk = 0; kk < 4; ++kk)
        af[kk] = *reinterpret_cast<const v2f*>(arow + 4 * kk + koff);
#pragma unroll
      for (int nt = 0; nt < 4; ++nt) {
        v8f c = {0.f, 0.f, 0.f, 0.f, 0.f, 0.f, 0.f, 0.f};
        const int col = nt * 16 + l16;
#pragma unroll
        for (int kk = 0; kk < 4; ++kk) {
          const int r = 4 * kk + koff;
          v2f b;
          b.x = W1[r * 64 + col];
          b.y = W1[(r + 1) * 64 + col];
          c = wmma4(af[kk], b, c);
        }
#pragma unroll
        for (int v = 0; v < 8; ++v) {
          // / sqrt(EDGE_ATTR_DIM) then silu, via TRANS exp+rcp (no IEEE div)
          Hs[(mt * 16 + v + 8 * half) * HS_LD + col] = fast_silu(c[v] * 0.25f);
        }
      }
    }

    // ---- GEMM2 per path (w = H @ W2 / 8) + tensor product -----------------
    for (int p = 0; p < 5; ++p) {
#pragma unroll
      for (int mt = 0; mt < 2; ++mt) {
        const float* hrow = Hs + (mt * 16 + l16) * HS_LD;
#pragma unroll
        for (int nt = 0; nt < 4; ++nt) {
          v8f c = {0.f, 0.f, 0.f, 0.f, 0.f, 0.f, 0.f, 0.f};
          const int col = p * 64 + nt * 16 + l16;
#pragma unroll
          for (int kk = 0; kk < 16; ++kk) {
            v2f a = *reinterpret_cast<const v2f*>(hrow + 4 * kk + koff);
            const int r = 4 * kk + koff;
            v2f b;
            b.x = W2[r * 320 + col];
            b.y = W2[(r + 1) * 320 + col];
            c = wmma4(a, b, c);
          }
#pragma unroll
          for (int v = 0; v < 8; ++v)
            wb[(mt * 16 + v + 8 * half) * WB_LD + nt * 16 + l16] = c[v] * 0.125f;
        }
      }

      // -- per-lane TP: w row of this edge lives at wb[lane][m*8+n] --------
      const float* wr = wb + lane * WB_LD;
      if (p == 0 || p == 1) {  // scalar input xs
        float y[8];
#pragma unroll
        for (int n = 0; n < 8; ++n) {
          float a = 0.f;
#pragma unroll
          for (int m = 0; m < 8; ++m) a += wr[m * 8 + n] * xs[m];
          y[n] = a;
        }
        if (p == 0) {
#pragma unroll
          for (int n = 0; n < 8; ++n) ts[phase][n] += y[n] * s0;
        } else {
#pragma unroll
          for (int n = 0; n < 8; ++n) {
            tvv[phase][n][0] += y[n] * s1x;
            tvv[phase][n][1] += y[n] * s1y;
            tvv[phase][n][2] += y[n] * s1z;
          }
        }
      } else if (p == 3) {  // scalar input dot(xv, s1)/sqrt3
        float dm[8];
#pragma unroll
        for (int m = 0; m < 8; ++m)
          dm[m] = (xv[m][0] * s1x + xv[m][1] * s1y + xv[m][2] * s1z) *
                  0.5773502691896258f;
#pragma unroll
        for (int n = 0; n < 8; ++n) {
          float a = 0.f;
#pragma unroll
          for (int m = 0; m < 8; ++m) a += wr[m * 8 + n] * dm[m];
          ts[phase][n] += a;
        }
      } else if (p == 2) {  // vector input xv, scaled by s0
#pragma unroll
        for (int n = 0; n < 8; ++n) {
          float a0 = 0.f, a1 = 0.f, a2 = 0.f;
#pragma unroll
          for (int m = 0; m < 8; ++m) {
            const float w = wr[m * 8 + n];
            a0 += w * xv[m][0]; a1 += w * xv[m][1]; a2 += w * xv[m][2];
          }
          tvv[phase][n][0] += a0 * s0;
          tvv[phase][n][1] += a1 * s0;
          tvv[phase][n][2] += a2 * s0;
        }
      } else {  // p == 4: vector input cross(xv, s1)/sqrt2
        float cr[8][3];
#pragma unroll
        for (int m = 0; m < 8; ++m) {
          cr[m][0] = (xv[m][1] * s1z - xv[m][2] * s1y) * 0.7071067811865476f;
          cr[m][1] = (xv[m][2] * s1x - xv[m][0] * s1z) * 0.7071067811865476f;
          cr[m][2] = (xv[m][0] * s1y - xv[m][1] * s1x) * 0.7071067811865476f;
        }
#pragma unroll
        for (int n = 0; n < 8; ++n) {
          float a0 = 0.f, a1 = 0.f, a2 = 0.f;
#pragma unroll
          for (int m = 0; m < 8; ++m) {
            const float w = wr[m * 8 + n];
            a0 += w * cr[m][0]; a1 += w * cr[m][1]; a2 += w * cr[m][2];
          }
          tvv[phase][n][0] += a0;
          tvv[phase][n][1] += a1;
          tvv[phase][n][2] += a2;
        }
      }
    }
  }

  // ---- finalize scales ----------------------------------------------------
  const float svS = 0.25f;                 // 1/sqrt(2*MUL)
  const float svV = 0.2041241452319315f;   // 1/sqrt(3*MUL)
  float ksn[8], vsn[8], kvn[8][3], vvn[8][3];
#pragma unroll
  for (int n = 0; n < 8; ++n) {
    ksn[n] = ts[0][n] * svS;
    vsn[n] = ts[1][n] * svS;
#pragma unroll
    for (int c = 0; c < 3; ++c) {
      kvn[n][c] = tvv[0][n][c] * svV;
      vvn[n][c] = tvv[1][n][c] * svV;
    }
  }

  // ---- logits, exp, segment accumulators ----------------------------------
  float l0 = 0.f, l1 = 0.f;
#pragma unroll
  for (int m = 0; m < 8; ++m) {
    const float qm = qs[dst * 8 + m];
#pragma unroll
    for (int n = 0; n < 8; ++n) l0 += Wd0[m * 8 + n] * qm * ksn[n];
  }
#pragma unroll
  for (int m = 0; m < 8; ++m) {
    const float q0 = qv[dst * 24 + m * 3 + 0];
    const float q1 = qv[dst * 24 + m * 3 + 1];
    const float q2 = qv[dst * 24 + m * 3 + 2];
#pragma unroll
    for (int n = 0; n < 8; ++n) {
      const float wd = Wd1[m * 8 + n];
      l1 += wd * (q0 * kvn[n][0] + q1 * kvn[n][1] + q2 * kvn[n][2]);
    }
  }
  const float logit = (l0 + l1 * 0.5773502691896258f) * 0.25f;
  const float ex = __expf(logit);
  expbuf[eL] = ex;
  atomicAdd(&zsum[dst], ex);
  atomicAdd(&cnt[dst], 1.0f);

  float* ve = vedge + (size_t)eL * 32;
#pragma unroll
  for (int n = 0; n < 8; ++n) ve[n] = vsn[n];
#pragma unroll
  for (int n = 0; n < 8; ++n) {
    ve[8 + n * 3 + 0] = vvn[n][0];
    ve[8 + n * 3 + 1] = vvn[n][1];
    ve[8 + n * 3 + 2] = vvn[n][2];
  }
}

// ---------------------------------------------------------------------------
__global__ void eqa_z_kernel(float* __restrict__ zsum, const float* __restrict__ cnt) {
  int i = blockIdx.x * blockDim.x + threadIdx.x;
  if (i >= kNodes) return;
  zsum[i] = zsum[i] * fast_rcp(fmaxf(cnt[i], 1.0f));
}

// ---------------------------------------------------------------------------
__global__ void eqa_scatter_kernel(const int* __restrict__ edge_index,
                                   const float* __restrict__ expbuf,
                                   const float* __restrict__ z,
                                   const float* __restrict__ vedge,
                                   float* __restrict__ out) {
  int e = blockIdx.x * blockDim.x + threadIdx.x;
  if (e >= kEdges) return;
  const int dst = edge_index[kEdges + e];
  const float a = expbuf[e] * fast_rcp(z[dst]);
  const float attn = sqrtf(fmaxf(a, 0.f));
  const float* ve = vedge + (size_t)e * 32;
  float* o = out + dst * 32;
#pragma unroll
  for (int i = 0; i < 32; ++i) atomicAdd(o + i, attn * ve[i]);
}

// ---------------------------------------------------------------------------
extern "C" void kernel_launch(void* const* d_in, const int* in_sizes, int n_in,
                              void* d_out, int out_size, void* d_ws, size_t ws_size,
                              hipStream_t stream) {
  const float* node_attr = (const float*)d_in[0];
  const int* edge_index = (const int*)d_in[1];
  const float* edge_attr = (const float*)d_in[2];
  const float* edge_sh = (const float*)d_in[3];
  const float* Wq0 = (const float*)d_in[4];
  const float* Wq1 = (const float*)d_in[5];
  const float* kW1 = (const float*)d_in[6];
  const float* kW2 = (const float*)d_in[7];
  const float* vW1 = (const float*)d_in[8];
  const float* vW2 = (const float*)d_in[9];
  const float* Wd0 = (const float*)d_in[10];
  const float* Wd1 = (const float*)d_in[11];
  float* out = (float*)d_out;

  // Workspace layout (floats): ~29.8 MB total.
  float* ws = (float*)d_ws;
  float* qs = ws;                       // kNodes*8
  float* qv = qs + (size_t)kNodes * 8;  // kNodes*24
  float* zsum = qv + (size_t)kNodes * 24;  // kNodes
  float* cnt = zsum + kNodes;              // kNodes (contiguous with zsum)
  float* expb = cnt + kNodes;              // kEdges
  float* vedge = expb + kEdges;            // kEdges*32

  eqa_zero_kernel<<<(kNodes * 32 + 255) / 256, 256, 0, stream>>>(out, zsum);
  eqa_q_kernel<<<(kNodes + 255) / 256, 256, 0, stream>>>(node_attr, Wq0, Wq1, qs, qv);

  const int tiles = kEdges / 32;  // 6250 wave-tiles
  const int blocksB = (tiles + WAVES_B - 1) / WAVES_B;
  const size_t smem = (size_t)WAVES_B * (32 * HS_LD + 32 * WB_LD) * sizeof(float);
  eqa_edge_kernel<<<blocksB, WAVES_B * 32, smem, stream>>>(
      node_attr, edge_index, edge_attr, edge_sh, kW1, kW2, vW1, vW2, Wd0, Wd1,
      qs, qv, zsum, cnt, expb, vedge);

  eqa_z_kernel<<<(kNodes + 255) / 256, 256, 0, stream>>>(zsum, cnt);
  eqa_scatter_kernel<<<(kEdges + 255) / 256, 256, 0, stream>>>(edge_index, expb,
                                                               zsum, vedge, out);
  (void)in_sizes; (void)n_in; (void)out_size; (void)ws_size;
}